// SGC_5643587027282
// MI455X (gfx1250) — compile-verified
//
#include <hip/hip_runtime.h>
#include <hip/hip_bf16.h>

// ---------------------------------------------------------------------------
// SGC: 3-hop normalized adjacency propagation + fused dual-linear head.
//   deg/dinv -> 3x (init + edge-scatter with native f32 atomics) ->
//   W_fused = W_sgc @ W_head, b_fused = b_sgc @ W_head + b_head ->
//   out = X3 @ W_fused + b_fused  via V_WMMA_F32_16X16X4_F32 (wave32).
// ---------------------------------------------------------------------------

typedef __attribute__((ext_vector_type(2))) float v2f;
typedef __attribute__((ext_vector_type(8))) float v8f;

#define DIM_IN 64
#define DIM_OUT 40
#define WPAD 48   // DIM_OUT padded to 3 WMMA N-tiles of 16

// ---- degree / normalization ------------------------------------------------

__global__ void deg_init_kernel(float* __restrict__ deg, int n) {
    int i = blockIdx.x * blockDim.x + threadIdx.x;
    if (i < n) deg[i] = 1.0f;                 // self-loop contributes 1
}

__global__ void deg_accum_kernel(const int* __restrict__ dst,
                                 float* __restrict__ deg, int n_edges) {
    int e = blockIdx.x * blockDim.x + threadIdx.x;
    if (e < n_edges) unsafeAtomicAdd(&deg[dst[e]], 1.0f);
}

__global__ void dinv_kernel(float* __restrict__ deg, int n) {
    int i = blockIdx.x * blockDim.x + threadIdx.x;
    if (i < n) deg[i] = rsqrtf(deg[i]);       // deg >= 1 always (self-loop)
}

// ---- one propagation hop ---------------------------------------------------

// y[i,:] = dinv[i]^2 * x[i,:]   (self-loop term; also serves as the zero-init)
__global__ void hop_init_kernel(const float* __restrict__ xin,
                                const float* __restrict__ dinv,
                                float* __restrict__ y, int n) {
    int i = blockIdx.x * blockDim.x + threadIdx.x;
    int total = n * DIM_IN;
    if (i < total) {
        int node = i >> 6;                    // DIM_IN == 64
        float d = dinv[node];
        y[i] = d * d * xin[i];
    }
}

// one wave (32 lanes) per edge: lane handles features {lane, lane+32}
__global__ __launch_bounds__(256) void hop_scatter_kernel(
    const float* __restrict__ xin, const float* __restrict__ dinv,
    const int* __restrict__ src, const int* __restrict__ dst,
    float* __restrict__ y, int n_edges) {
    int lane = threadIdx.x & 31;
    int e = blockIdx.x * (blockDim.x >> 5) + (threadIdx.x >> 5);
    if (e >= n_edges) return;
    int s = src[e];
    int d = dst[e];
    float w = dinv[s] * dinv[d];
    const float* xi = xin + (size_t)s * DIM_IN;
    float* yo = y + (size_t)d * DIM_IN;
    unsafeAtomicAdd(&yo[lane],      w * xi[lane]);
    unsafeAtomicAdd(&yo[lane + 32], w * xi[lane + 32]);
}

// ---- fuse the two linears: Wf = Wsgc @ Whead, bf = bsgc @ Whead + bhead ----

__global__ void fuse_weights_kernel(const float* __restrict__ Wsgc,   // [64,64]
                                    const float* __restrict__ bsgc,   // [64]
                                    const float* __restrict__ Whead,  // [64,40]
                                    const float* __restrict__ bhead,  // [40]
                                    float* __restrict__ Wf,           // [64,48]
                                    float* __restrict__ bf) {         // [48]
    int n = threadIdx.x;      // 0..47
    int k = blockIdx.x;       // 0..63 weight rows, 64 => bias row
    if (k < DIM_IN) {
        float s = 0.0f;
        if (n < DIM_OUT) {
            for (int j = 0; j < DIM_IN; ++j)
                s += Wsgc[k * DIM_IN + j] * Whead[j * DIM_OUT + n];
        }
        Wf[k * WPAD + n] = s;                 // cols 40..47 stay zero
    } else {
        float s = 0.0f;
        if (n < DIM_OUT) {
            for (int j = 0; j < DIM_IN; ++j)
                s += bsgc[j] * Whead[j * DIM_OUT + n];
            s += bhead[n];
        }
        bf[n] = s;
    }
}

// ---- WMMA f32 GEMM head: out[N,40] = X[N,64] @ Wf[64,48] + bf --------------
// One wave per 16-node M-tile; 3 N-tiles of 16; K=64 as 16 steps of K=4.

__global__ __launch_bounds__(256) void sgc_head_gemm_kernel(
    const float* __restrict__ X,     // [N,64]
    const float* __restrict__ Wf,    // [64,48] row-major, cols >=40 zero
    const float* __restrict__ bf,    // [48]
    float* __restrict__ out,         // [N,40]
    int n_nodes) {
    const int lane = threadIdx.x & 31;
    const int wave = threadIdx.x >> 5;
    const int tile = blockIdx.x * 8 + wave;       // M-tile index
    const int m_base = tile * 16;
    if (m_base >= n_nodes) return;

    // A frag (16x4 f32): lanes 0-15 -> row=lane, K=k0+{0,1};
    //                    lanes 16-31 -> row=lane-16, K=k0+{2,3}
    const int arow = m_base + (lane & 15);
    const int kofs = (lane >> 4) * 2;             // 0 or 2
    const int bn   = lane & 15;

    v8f acc0 = {}, acc1 = {}, acc2 = {};
    #pragma unroll
    for (int k0 = 0; k0 < DIM_IN; k0 += 4) {
        const float* ap = X + (size_t)arow * DIM_IN + k0 + kofs;
        v2f a;
        a.x = ap[0];
        a.y = ap[1];
        const int bk = k0 + kofs;                 // rows of B for this lane group
        v2f b0, b1, b2;
        b0.x = Wf[bk * WPAD + bn];          b0.y = Wf[(bk + 1) * WPAD + bn];
        b1.x = Wf[bk * WPAD + 16 + bn];     b1.y = Wf[(bk + 1) * WPAD + 16 + bn];
        b2.x = Wf[bk * WPAD + 32 + bn];     b2.y = Wf[(bk + 1) * WPAD + 32 + bn];
        acc0 = __builtin_amdgcn_wmma_f32_16x16x4_f32(false, a, false, b0,
                                                     (short)0, acc0, false, false);
        acc1 = __builtin_amdgcn_wmma_f32_16x16x4_f32(false, a, false, b1,
                                                     (short)0, acc1, false, false);
        acc2 = __builtin_amdgcn_wmma_f32_16x16x4_f32(false, a, false, b2,
                                                     (short)0, acc2, false, false);
    }

    // D layout: VGPR v, lanes 0-15 -> row m_base+v, lanes 16-31 -> row m_base+v+8
    const int colg = lane & 15;
    const int radd = (lane >> 4) * 8;
    const float bia0 = bf[colg];
    const float bia1 = bf[16 + colg];
    const float bia2 = (colg < 8) ? bf[32 + colg] : 0.0f;
    #pragma unroll
    for (int v = 0; v < 8; ++v) {
        const int row = m_base + v + radd;
        float* orow = out + (size_t)row * DIM_OUT;
        orow[colg]      = acc0[v] + bia0;
        orow[16 + colg] = acc1[v] + bia1;
        if (colg < 8) orow[32 + colg] = acc2[v] + bia2;   // cols 32..39 only
    }
}

// ---------------------------------------------------------------------------

extern "C" void kernel_launch(void* const* d_in, const int* in_sizes, int n_in,
                              void* d_out, int out_size, void* d_ws, size_t ws_size,
                              hipStream_t stream) {
    const float* x     = (const float*)d_in[0];
    const int*   ei    = (const int*)d_in[1];   // [2, E] (JAX default int32)
    const float* Wsgc  = (const float*)d_in[2];
    const float* bsgc  = (const float*)d_in[3];
    const float* Whead = (const float*)d_in[4];
    const float* bhead = (const float*)d_in[5];
    float* out = (float*)d_out;

    const int N = in_sizes[0] / DIM_IN;         // 100000
    const int E = in_sizes[1] / 2;              // 1600000
    const int* src = ei;
    const int* dst = ei + E;

    // workspace carve-out (256B aligned slices)
    char* ws = (char*)d_ws;
    size_t off = 0;
    auto take = [&](size_t bytes) -> void* {
        void* p = ws + off;
        off += (bytes + 255) & ~(size_t)255;
        return p;
    };
    float* dinv = (float*)take((size_t)N * sizeof(float));
    float* buf0 = (float*)take((size_t)N * DIM_IN * sizeof(float));
    float* buf1 = (float*)take((size_t)N * DIM_IN * sizeof(float));
    float* Wf   = (float*)take((size_t)DIM_IN * WPAD * sizeof(float));
    float* bf   = (float*)take((size_t)WPAD * sizeof(float));

    const int T = 256;
    // degree + normalization (dinv buffer doubles as deg)
    deg_init_kernel<<<(N + T - 1) / T, T, 0, stream>>>(dinv, N);
    deg_accum_kernel<<<(E + T - 1) / T, T, 0, stream>>>(dst, dinv, E);
    dinv_kernel<<<(N + T - 1) / T, T, 0, stream>>>(dinv, N);

    // fused head weights (independent of hops; overlaps fine on stream order)
    fuse_weights_kernel<<<DIM_IN + 1, WPAD, 0, stream>>>(Wsgc, bsgc, Whead, bhead, Wf, bf);

    const int totF = N * DIM_IN;
    const int gInit = (totF + T - 1) / T;
    const int gScat = (E + 7) / 8;              // 8 waves/block, 1 edge/wave

    // hop 1: x -> buf0
    hop_init_kernel<<<gInit, T, 0, stream>>>(x, dinv, buf0, N);
    hop_scatter_kernel<<<gScat, T, 0, stream>>>(x, dinv, src, dst, buf0, E);
    // hop 2: buf0 -> buf1
    hop_init_kernel<<<gInit, T, 0, stream>>>(buf0, dinv, buf1, N);
    hop_scatter_kernel<<<gScat, T, 0, stream>>>(buf0, dinv, src, dst, buf1, E);
    // hop 3: buf1 -> buf0
    hop_init_kernel<<<gInit, T, 0, stream>>>(buf1, dinv, buf0, N);
    hop_scatter_kernel<<<gScat, T, 0, stream>>>(buf1, dinv, src, dst, buf0, E);

    // WMMA head GEMM: out = buf0 @ Wf + bf
    const int tiles = (N + 15) / 16;
    sgc_head_gemm_kernel<<<(tiles + 7) / 8, T, 0, stream>>>(buf0, Wf, bf, out, N);
}